// GraphKANLayer_51367808860745
// MI455X (gfx1250) — compile-verified
//
#include <hip/hip_runtime.h>
#include <hip/hip_bf16.h>

// GraphKAN layer fused kernel for MI455X (gfx1250, wave32, WMMA).
// agg = adj @ x  (per batch, M=8192, N=64x4, K=8192)  -> bf16 WMMA, f32 accum
// agg_nl = agg * (inv_sqrt2 * nl_w) + nl_b            -> scale folded into x pack
// out = agg_nl @ w_out^T + b_out                      -> small VALU epilogue from LDS

#define NODE 8192
#define FEAT 64
#define OUTD 64
#define BATCH 4
#define NTOT (BATCH * FEAT) /* 256 */
#define MT 32               /* output rows per workgroup */
#define KSTEP 64            /* K advanced per LDS stage (2 wmma chunks) */
#define ABUF 4096           /* bytes per A-tile LDS buffer (2 chunks x 2 KB) */
#define INV_SQRT2 0.70710678118654752f

typedef __attribute__((ext_vector_type(16))) __bf16 v16bf;
typedef __attribute__((ext_vector_type(8)))  float  v8f;
typedef __attribute__((ext_vector_type(4)))  unsigned int ui4;

// Scheduling fence: keep the next chunk's fragment loads from being hoisted
// over this point. That bounds concurrent A/B tuple liveness, keeping VGPR
// pressure under the 4-waves/SIMD budget so the allocator never splits the
// accumulator live ranges (which showed up as v_mov_b64 rotation copies).
#if __has_builtin(__builtin_amdgcn_sched_barrier)
#define SCHED_FENCE() __builtin_amdgcn_sched_barrier(0)
#else
#define SCHED_FENCE() asm volatile("" ::: "memory")
#endif

// One k-chunk of the 2x2 tile: 4 back-to-back WMMAs with the accumulator
// TIED (D==C via "+v"), so accumulators cannot be renamed across the chain.
// Trailing s_nop 4 covers the bf16 WMMA->VALU hazard window for any compiler
// VALU scheduled after the block (epilogue reads, conversion code).
__device__ __forceinline__ void wmma_group(v8f& c00, v8f& c01, v8f& c10, v8f& c11,
                                           v16bf a0, v16bf a1, v16bf b0, v16bf b1) {
  asm volatile(
      "v_wmma_f32_16x16x32_bf16 %0, %4, %6, %0\n\t"
      "v_wmma_f32_16x16x32_bf16 %1, %4, %7, %1\n\t"
      "v_wmma_f32_16x16x32_bf16 %2, %5, %6, %2\n\t"
      "v_wmma_f32_16x16x32_bf16 %3, %5, %7, %3\n\t"
      "s_nop 4"
      : "+v"(c00), "+v"(c01), "+v"(c10), "+v"(c11)
      : "v"(a0), "v"(a1), "v"(b0), "v"(b1));
}

// f32 -> bf16 pair, round-half-up (one v_add per input + one v_perm to pack).
__device__ __forceinline__ unsigned int f2bf_pk(float lo, float hi) {
  const unsigned a = __float_as_uint(lo) + 0x8000u;
  const unsigned b = __float_as_uint(hi) + 0x8000u;
#if __has_builtin(__builtin_amdgcn_perm)
  return __builtin_amdgcn_perm(b, a, 0x07060302u); // {b.hi16, a.hi16}
#else
  return (a >> 16) | (b & 0xFFFF0000u);
#endif
}

__device__ __forceinline__ unsigned short f2bf(float f) {
  return (unsigned short)((__float_as_uint(f) + 0x8000u) >> 16);
}

// ---------------------------------------------------------------------------
// Kernel 1: pack x (4,8192,64) f32 -> xT[n=b*64+f][j] bf16, pre-scaled by
// inv_sqrt2 * nl_w[f]. 4 MiB panel, stays hot in the 192 MB L2.
// ---------------------------------------------------------------------------
__global__ void __launch_bounds__(256)
pack_x_kernel(const float* __restrict__ x, const float* __restrict__ nl_w,
              unsigned short* __restrict__ xT) {
  const int idx = blockIdx.x * 256 + threadIdx.x; // 0 .. NTOT*NODE-1
  const int j = idx & (NODE - 1);                 // node index (coalesced writes)
  const int n = idx >> 13;                        // NODE == 2^13
  const int b = n >> 6, f = n & 63;
  const float v = x[((size_t)(b * NODE + j)) * FEAT + f] * (INV_SQRT2 * nl_w[f]);
  xT[(size_t)n * NODE + j] = f2bf(v);
}

// ---------------------------------------------------------------------------
// Kernel 2: fused GEMM + epilogue. One WG = 32 rows x all 256 (batch,feat)
// columns, so each adj row-strip is streamed from HBM exactly once.
// A tile double-buffered in LDS; last K-step peeled so the steady-state loop
// is branch-free.
// ---------------------------------------------------------------------------
__global__ void __launch_bounds__(256, 1)
gkan_fused_kernel(const float* __restrict__ adj,
                  const unsigned short* __restrict__ xT,
                  const float* __restrict__ nl_b,
                  const float* __restrict__ w_out,
                  const float* __restrict__ b_out,
                  float* __restrict__ out) {
  __shared__ float s_w[FEAT * OUTD];   // 16 KB, w_out transposed: s_w[f*64+o]
  __shared__ float s_agg[MT * NTOT];   // 32 KB; first 8 KB aliased as 2 bf16 A bufs

  const int tid = threadIdx.x;
  const int lane = tid & 31;
  const int wid = tid >> 5;            // 8 waves
  const int m0 = blockIdx.x * MT;

  // Stage w_out transposed for conflict-free epilogue reads (lanes vary o).
  for (int i = tid; i < FEAT * OUTD; i += 256) {
    const int o = i >> 6, f = i & 63;
    s_w[f * OUTD + o] = w_out[i];
  }

  // ---- A staging geometry: thread t loads 8 consecutive f32 of adj and
  // stores one 16B bf16 chunk in WMMA A-fragment layout (ISA 7.12.2):
  //   lane L<16  : row L, K {0..7 | half0, 16..23 | half1}
  //   lane 16+L  : row L, K {8..15 | half0, 24..31 | half1}
  const int sr = tid >> 3;             // row 0..31
  const int sq = tid & 7;              // K octet 0..7 within KSTEP
  const int s_kc = sq >> 2;            // which 32-wide wmma chunk
  const int s_j = sq & 3;              // octet within chunk
  char* s_a = (char*)s_agg;            // 8 KB bf16 A area (aliased)
  const int s_off = s_kc * 2048 + (sr >> 4) * 1024 +
                    (((sr & 15) + 16 * (s_j & 1)) * 32) + (s_j >> 1) * 16;
  const float* aRow = adj + (size_t)(m0 + sr) * NODE + sq * 8;

  // ---- B fragment base pointers: lane L reads 16 contiguous bf16 of column
  // n = wid*32 + nt*16 + (L&15), K sub-range selected by (L>>4)*16.
  const unsigned short* xrow0 =
      xT + (size_t)(wid * 32 + (lane & 15)) * NODE + 16 * (lane >> 4);
  const unsigned short* xrow1 = xrow0 + (size_t)16 * NODE;

  v8f acc00 = {}, acc01 = {}, acc10 = {}, acc11 = {}; // [m-tile][n-tile]

  // Prologue: stage first A tile into buffer 0.
  {
    const float4 a0 = *(const float4*)(aRow);
    const float4 a1 = *(const float4*)(aRow + 4);
    ui4 p;
    p.x = f2bf_pk(a0.x, a0.y);
    p.y = f2bf_pk(a0.z, a0.w);
    p.z = f2bf_pk(a1.x, a1.y);
    p.w = f2bf_pk(a1.z, a1.w);
    *(ui4*)(s_a + s_off) = p;
  }

  int bufoff = 0;
  // Steady state: branch-free; stages k0+KSTEP while computing k0.
  for (int k0 = 0; k0 < NODE - KSTEP; k0 += KSTEP) {
    __syncthreads(); // current buffer staged; other buffer's readers done

    // Next A tile (HBM) — issued first so it runs under the WMMAs.
    const float4 a0 = *(const float4*)(aRow + k0 + KSTEP);
    const float4 a1 = *(const float4*)(aRow + k0 + KSTEP + 4);

#pragma unroll
    for (int kc = 0; kc < 2; ++kc) {
      const int kk = k0 + kc * 32;
      const v16bf b0 = *(const v16bf*)(xrow0 + kk);
      const v16bf b1 = *(const v16bf*)(xrow1 + kk);
      const v16bf af0 = *(const v16bf*)(s_a + bufoff + kc * 2048 + lane * 32);
      const v16bf af1 = *(const v16bf*)(s_a + bufoff + kc * 2048 + 1024 + lane * 32);
      wmma_group(acc00, acc01, acc10, acc11, af0, af1, b0, b1);
      SCHED_FENCE();
    }

    // Convert + store next tile into the other buffer (safe after the
    // barrier above: all reads of that buffer finished last iteration).
    ui4 p;
    p.x = f2bf_pk(a0.x, a0.y);
    p.y = f2bf_pk(a0.z, a0.w);
    p.z = f2bf_pk(a1.x, a1.y);
    p.w = f2bf_pk(a1.z, a1.w);
    *(ui4*)(s_a + (bufoff ^ ABUF) + s_off) = p;

    bufoff ^= ABUF;
  }

  // Peeled final K-step (no staging, no branches in the loop above).
  {
    const int k0 = NODE - KSTEP;
    __syncthreads();
#pragma unroll
    for (int kc = 0; kc < 2; ++kc) {
      const int kk = k0 + kc * 32;
      const v16bf b0 = *(const v16bf*)(xrow0 + kk);
      const v16bf b1 = *(const v16bf*)(xrow1 + kk);
      const v16bf af0 = *(const v16bf*)(s_a + bufoff + kc * 2048 + lane * 32);
      const v16bf af1 = *(const v16bf*)(s_a + bufoff + kc * 2048 + 1024 + lane * 32);
      wmma_group(acc00, acc01, acc10, acc11, af0, af1, b0, b1);
      SCHED_FENCE();
    }
  }

  // ---- Epilogue 1: agg + nl_b -> LDS f32 [m][n]. C layout (ISA 7.12.2):
  // lane<16: VGPR v = row v; lane>=16: VGPR v = row 8+v; N = lane&15.
  __syncthreads();
  const int ncol = wid * 32 + (lane & 15);
  const float nlb0 = nl_b[ncol & 63];
  const float nlb1 = nl_b[(ncol + 16) & 63];
  const int mrowbase = 8 * (lane >> 4);
#pragma unroll
  for (int v = 0; v < 8; ++v) {
    s_agg[(mrowbase + v) * NTOT + ncol]             = acc00[v] + nlb0;
    s_agg[(mrowbase + v) * NTOT + ncol + 16]        = acc01[v] + nlb1;
    s_agg[(16 + mrowbase + v) * NTOT + ncol]        = acc10[v] + nlb0;
    s_agg[(16 + mrowbase + v) * NTOT + ncol + 16]   = acc11[v] + nlb1;
  }
  __syncthreads();

  // ---- Epilogue 2: out[b, m0+m, o] = sum_f agg_nl[m, b*64+f]*w_out[o,f] + b_out[o]
  // Thread -> (b, o); agg reads are wave-broadcast, w reads conflict-free.
  const int o = tid & 63;
  const int b = tid >> 6;
  float wcol[FEAT];
#pragma unroll
  for (int f = 0; f < FEAT; ++f) wcol[f] = s_w[f * OUTD + o];
  const float bo = b_out[o];
  for (int m = 0; m < MT; ++m) {
    const float* row = s_agg + m * NTOT + b * FEAT;
    float a2 = bo;
#pragma unroll
    for (int f = 0; f < FEAT; ++f) a2 += row[f] * wcol[f];
    out[((size_t)b * NODE + (m0 + m)) * OUTD + o] = a2;
  }
}

extern "C" void kernel_launch(void* const* d_in, const int* in_sizes, int n_in,
                              void* d_out, int out_size, void* d_ws, size_t ws_size,
                              hipStream_t stream) {
  const float* x     = (const float*)d_in[0];
  const float* adj   = (const float*)d_in[1];
  const float* nl_w  = (const float*)d_in[2];
  const float* nl_b  = (const float*)d_in[3];
  const float* w_out = (const float*)d_in[4];
  const float* b_out = (const float*)d_in[5];
  float* out = (float*)d_out;
  unsigned short* xT = (unsigned short*)d_ws; // 256*8192*2 = 4 MiB scratch

  pack_x_kernel<<<(NTOT * NODE) / 256, 256, 0, stream>>>(x, nl_w, xT);
  gkan_fused_kernel<<<NODE / MT, 256, 0, stream>>>(adj, xT, nl_b, w_out, b_out, out);

  (void)in_sizes; (void)n_in; (void)out_size; (void)ws_size;
}